// Volume_89034672046121
// MI455X (gfx1250) — compile-verified
//
#include <hip/hip_runtime.h>

// Trilinear interpolation: 4,194,304 points into a 512^3 fp32 volume.
// Latency-bound random gather. MI455X strategy:
//  - 2 points per lane -> ~16 independent corner gathers in flight per lane
//    (plus 2 coord loads), doubling per-wave MLP at trivial VGPR cost
//  - non-temporal hints (TH=NT) on the once-streamed coords/output so the
//    192MB L2 stays dedicated to the 512MB volume
//  - wave32-friendly 256-thread blocks

#define VD 512
#define VH 512
#define VW 512

struct Corners {
    const float* r00;
    const float* r01;
    const float* r10;
    const float* r11;
    int x0, x1;
    float wz, wy, wx;
};

__device__ __forceinline__ Corners make_corners(const float* __restrict__ vol,
                                                float cz, float cy, float cx)
{
    const float S = 511.0f;
    float sz = cz * S, sy = cy * S, sx = cx * S;

    // Reference semantics: truncate-to-int, weights from UNclamped index.
    int iz = (int)sz, iy = (int)sy, ix = (int)sx;

    Corners c;
    c.wz = sz - (float)iz;
    c.wy = sy - (float)iy;
    c.wx = sx - (float)ix;

    int z0 = min(max(iz, 0), VD - 1);
    int y0 = min(max(iy, 0), VH - 1);
    c.x0  = min(max(ix, 0), VW - 1);
    int z1 = min(z0 + 1, VD - 1);
    int y1 = min(y0 + 1, VH - 1);
    c.x1  = min(c.x0 + 1, VW - 1);

    int zb0 = z0 << 18, zb1 = z1 << 18;   // z * 512 * 512
    int yb0 = y0 << 9,  yb1 = y1 << 9;    // y * 512
    c.r00 = vol + (zb0 + yb0);
    c.r01 = vol + (zb0 + yb1);
    c.r10 = vol + (zb1 + yb0);
    c.r11 = vol + (zb1 + yb1);
    return c;
}

// 2 points per thread: block handles 512 consecutive points.
__global__ __launch_bounds__(256) void trilerp2_kernel(
    const float* __restrict__ coords,
    const float* __restrict__ vol,
    float* __restrict__ out)
{
    int base = blockIdx.x * 512 + threadIdx.x;

    // Phase 1: both coord triples (coalesced b96, NT).
    float cz[2], cy[2], cx[2];
#pragma unroll
    for (int k = 0; k < 2; ++k) {
        int i = base + k * 256;
        cz[k] = __builtin_nontemporal_load(&coords[3 * i + 0]);
        cy[k] = __builtin_nontemporal_load(&coords[3 * i + 1]);
        cx[k] = __builtin_nontemporal_load(&coords[3 * i + 2]);
    }

    // Phase 2: index math for both points.
    Corners cc[2];
#pragma unroll
    for (int k = 0; k < 2; ++k)
        cc[k] = make_corners(vol, cz[k], cy[k], cx[k]);

    // Phase 3: issue all 16 gathers before any consumer (deep MLP clause).
    float v[2][8];
#pragma unroll
    for (int k = 0; k < 2; ++k) {
        v[k][0] = cc[k].r00[cc[k].x0];
        v[k][1] = cc[k].r00[cc[k].x1];
        v[k][2] = cc[k].r01[cc[k].x0];
        v[k][3] = cc[k].r01[cc[k].x1];
        v[k][4] = cc[k].r10[cc[k].x0];
        v[k][5] = cc[k].r10[cc[k].x1];
        v[k][6] = cc[k].r11[cc[k].x0];
        v[k][7] = cc[k].r11[cc[k].x1];
    }

    // Phase 4: lerp chains + NT stores.  lerp(a,b,w) = fma(w, b-a, a)
#pragma unroll
    for (int k = 0; k < 2; ++k) {
        float wx = cc[k].wx, wy = cc[k].wy, wz = cc[k].wz;
        float c00 = fmaf(wx, v[k][1] - v[k][0], v[k][0]);
        float c01 = fmaf(wx, v[k][3] - v[k][2], v[k][2]);
        float c10 = fmaf(wx, v[k][5] - v[k][4], v[k][4]);
        float c11 = fmaf(wx, v[k][7] - v[k][6], v[k][6]);
        float c0  = fmaf(wy, c01 - c00, c00);
        float c1  = fmaf(wy, c11 - c10, c10);
        float r   = fmaf(wz, c1 - c0, c0);
        __builtin_nontemporal_store(r, &out[base + k * 256]);
    }
}

// Defensive tail (not used when n % 512 == 0, which holds for N = 2^22).
__global__ __launch_bounds__(256) void trilerp1_kernel(
    const float* __restrict__ coords,
    const float* __restrict__ vol,
    float* __restrict__ out,
    int first, int n)
{
    int i = first + blockIdx.x * 256 + threadIdx.x;
    if (i >= n) return;

    float cz = __builtin_nontemporal_load(&coords[3 * i + 0]);
    float cy = __builtin_nontemporal_load(&coords[3 * i + 1]);
    float cx = __builtin_nontemporal_load(&coords[3 * i + 2]);

    Corners c = make_corners(vol, cz, cy, cx);

    float c000 = c.r00[c.x0], c001 = c.r00[c.x1];
    float c010 = c.r01[c.x0], c011 = c.r01[c.x1];
    float c100 = c.r10[c.x0], c101 = c.r10[c.x1];
    float c110 = c.r11[c.x0], c111 = c.r11[c.x1];

    float c00 = fmaf(c.wx, c001 - c000, c000);
    float c01 = fmaf(c.wx, c011 - c010, c010);
    float c10 = fmaf(c.wx, c101 - c100, c100);
    float c11 = fmaf(c.wx, c111 - c110, c110);
    float c0  = fmaf(c.wy, c01 - c00, c00);
    float c1  = fmaf(c.wy, c11 - c10, c10);
    float r   = fmaf(c.wz, c1 - c0, c0);
    __builtin_nontemporal_store(r, &out[i]);
}

extern "C" void kernel_launch(void* const* d_in, const int* in_sizes, int n_in,
                              void* d_out, int out_size, void* d_ws, size_t ws_size,
                              hipStream_t stream)
{
    const float* coords = (const float*)d_in[0];  // N*3 floats
    const float* vol    = (const float*)d_in[1];  // 512^3 floats
    float* out          = (float*)d_out;          // N floats

    int n = in_sizes[0] / 3;                      // 4,194,304
    int full_blocks = n / 512;                    // 8192 for N = 2^22
    if (full_blocks > 0)
        trilerp2_kernel<<<full_blocks, 256, 0, stream>>>(coords, vol, out);

    int done = full_blocks * 512;
    int rem = n - done;
    if (rem > 0) {
        int tail_blocks = (rem + 255) / 256;
        trilerp1_kernel<<<tail_blocks, 256, 0, stream>>>(coords, vol, out, done, n);
    }
}